// MoonshineStreamingDecoderAttention_1133871366235
// MI455X (gfx1250) — compile-verified
//
#include <hip/hip_runtime.h>
#include <math.h>

// MI455X / gfx1250 decode-attention (GQA, Tq=1).
// Roofline: 168 MB fp32 KV stream @ 23.3 TB/s ~= 7.2 us floor; attention math
// (~336 MFLOP) is free; projections use V_WMMA_F32_16X16X4_F32 (fp32-exact).

typedef float v2f __attribute__((ext_vector_type(2)));
typedef float v8f __attribute__((ext_vector_type(8)));

#define NHQ   8
#define NKVH  2
#define HD    40
#define RD    32
#define BB    32
#define HDIM  320
#define TPAST 8192
#define TTOT  8193
#define NCHUNK 8
#define CHUNK 1024
#define PREC  42   // per-(b,head,chunk) partial record: m, l, acc[40]

// -------- fp32 WMMA 16x16 tile GEMM helper --------------------------------
// Computes D(16x16) = A(16xK) * B(KxN=16) with B[k,n] = W[wrow0+n][k].
// A-frag layout (ISA 7.12.2, 32-bit A 16x4): lanes 0-15 M=0..15 K={0,1},
// lanes 16-31 same M, K={2,3}. B-frag mirrors it (row-striped across lanes).
__device__ __forceinline__ void wmma_tile_f32(
    const float* __restrict__ A, int lda, int arow0,
    const float* __restrict__ W, int ldw, int wrow0,
    int K, float* c /*[8]*/)
{
  const int lane = threadIdx.x & 31;
  const int lm   = lane & 15;
  const int half = lane >> 4;
  v8f acc = {0.f,0.f,0.f,0.f,0.f,0.f,0.f,0.f};
  const float* arow = A + (size_t)(arow0 + lm) * lda + 2 * half;
  const float* wrow = W + (size_t)(wrow0 + lm) * ldw + 2 * half;
  for (int k = 0; k < K; k += 4) {
    v2f a = *(const v2f*)(arow + k);
    v2f b = *(const v2f*)(wrow + k);
    acc = __builtin_amdgcn_wmma_f32_16x16x4_f32(
        /*neg_a=*/false, a, /*neg_b=*/false, b,
        /*c_mod=*/(short)0, acc, /*reuse_a=*/false, /*reuse_b=*/false);
  }
#pragma unroll
  for (int r = 0; r < 8; ++r) c[r] = acc[r];
}

// -------- Kernel 1: fused QKV projection (WMMA) ---------------------------
// Output columns: [0,320) -> q, [320,400) -> k_new, [400,480) -> v_new.
__global__ void qkv_proj_kernel(const float* __restrict__ hs,
                                const float* __restrict__ Wq,
                                const float* __restrict__ Wk,
                                const float* __restrict__ Wv,
                                float* __restrict__ qbuf,
                                float* __restrict__ kbuf,
                                float* __restrict__ vbuf)
{
  const int tile  = blockIdx.x;          // 60 tiles: 2 (M) x 30 (N)
  const int tm    = tile & 1;
  const int tn    = tile >> 1;
  const int nbase = tn * 16;

  const float* W; int wrow0;
  if (nbase < 320)      { W = Wq; wrow0 = nbase; }
  else if (nbase < 400) { W = Wk; wrow0 = nbase - 320; }
  else                  { W = Wv; wrow0 = nbase - 400; }

  float c[8];
  wmma_tile_f32(hs, HDIM, tm * 16, W, HDIM, wrow0, HDIM, c);

  const int lane = threadIdx.x & 31;
  const int lm   = lane & 15;
  const int half = lane >> 4;
  const int o    = nbase + lm;
#pragma unroll
  for (int r = 0; r < 8; ++r) {
    const int m = tm * 16 + r + 8 * half;   // batch row
    if (o < 320)      qbuf[m * 320 + o]        = c[r];
    else if (o < 400) kbuf[m * 80 + (o - 320)] = c[r];
    else              vbuf[m * 80 + (o - 400)] = c[r];
  }
}

// -------- Kernel 2: RoPE on q and k_new (interleaved pairs) ---------------
__global__ void rope_kernel(const float* __restrict__ cosb,
                            const float* __restrict__ sinb,
                            float* __restrict__ qbuf,
                            float* __restrict__ kbuf)
{
  const int b = blockIdx.x;
  const int t = threadIdx.x;            // 0..159: (NHQ+NKVH)*RD/2 pairs
  const int unit = t >> 4;              // head unit (0..7 q, 8..9 k)
  const int i    = t & 15;              // pair index -> cos/sin[i]
  const float c = cosb[b * RD + i];
  const float s = sinb[b * RD + i];
  float* buf;
  if (unit < NHQ) buf = qbuf + b * 320 + unit * HD;
  else            buf = kbuf + b * 80 + (unit - NHQ) * HD;
  const int off = 2 * i;
  const float x0 = buf[off], x1 = buf[off + 1];
  buf[off]     = x0 * c - x1 * s;
  buf[off + 1] = x1 * c + x0 * s;
}

// -------- Kernel 3: flash-decode partial over KV cache --------------------
// grid = B*NKV*NCHUNK, block = 256 (8 waves). Lane-quad per key: 10 dims/lane.
// 4 query heads share each K/V row (GQA). Online softmax, then wave shfl
// reduction + LDS cross-wave reduction -> partial record per (b,head,chunk).
__global__ __launch_bounds__(256) void flash_partial_kernel(
    const float* __restrict__ kc, const float* __restrict__ vc,
    const float* __restrict__ maskp, const float* __restrict__ qbuf,
    float* __restrict__ partial)
{
  const int blk   = blockIdx.x;
  const int chunk = blk % NCHUNK;
  const int kv    = (blk / NCHUNK) % NKVH;
  const int b     = blk / (NCHUNK * NKVH);
  const int tid   = threadIdx.x;
  const int wave  = tid >> 5;
  const int lane  = tid & 31;
  const int kq    = lane >> 2;      // key-in-group 0..7
  const int part  = lane & 3;       // dim slice: [part*10, part*10+10)
  const float scale = 0.15811388300841897f; // 1/sqrt(40)

  float qh[4][10];
#pragma unroll
  for (int h = 0; h < 4; ++h)
#pragma unroll
    for (int j = 0; j < 10; ++j)
      qh[h][j] = qbuf[b * 320 + (kv * 4 + h) * HD + part * 10 + j];

  float m[4] = {-INFINITY, -INFINITY, -INFINITY, -INFINITY};
  float l[4] = {0.f, 0.f, 0.f, 0.f};
  float acc[4][10];
#pragma unroll
  for (int h = 0; h < 4; ++h)
#pragma unroll
    for (int j = 0; j < 10; ++j) acc[h][j] = 0.f;

  const float* kbase = kc + (size_t)(b * NKVH + kv) * TPAST * HD;
  const float* vbase = vc + (size_t)(b * NKVH + kv) * TPAST * HD;

  for (int it = 0; it < CHUNK / 64; ++it) {
    const int key = chunk * CHUNK + it * 64 + wave * 8 + kq;
    const float* krp = kbase + (size_t)key * HD + part * 10;
    const float* vrp = vbase + (size_t)key * HD + part * 10;
    float kd[10], vr[10];
#pragma unroll
    for (int j = 0; j < 5; ++j) {
      v2f k2 = *(const v2f*)(krp + 2 * j);
      kd[2 * j] = k2.x; kd[2 * j + 1] = k2.y;
      v2f vv2 = *(const v2f*)(vrp + 2 * j);
      vr[2 * j] = vv2.x; vr[2 * j + 1] = vv2.y;
    }
    const float mk = maskp[b * TTOT + key];

#pragma unroll
    for (int h = 0; h < 4; ++h) {
      float s = 0.f;
#pragma unroll
      for (int j = 0; j < 10; ++j) s += qh[h][j] * kd[j];
      // reduce over the lane-quad (dim slices)
      s += __shfl_xor(s, 1, 32);
      s += __shfl_xor(s, 2, 32);
      s = s * scale + mk;
      if (s > m[h]) {                 // conditional rescale (rare)
        const float f = __expf(m[h] - s);
        l[h] *= f;
#pragma unroll
        for (int j = 0; j < 10; ++j) acc[h][j] *= f;
        m[h] = s;
      }
      const float p = __expf(s - m[h]);
      l[h] += p;
#pragma unroll
      for (int j = 0; j < 10; ++j) acc[h][j] += p * vr[j];
    }
  }

  // reduce across key-groups within the wave (xor over kq bits: 4, 8, 16)
#pragma unroll
  for (int xm = 4; xm <= 16; xm <<= 1) {
#pragma unroll
    for (int h = 0; h < 4; ++h) {
      const float om = __shfl_xor(m[h], xm, 32);
      const float ol = __shfl_xor(l[h], xm, 32);
      float oa[10];
#pragma unroll
      for (int j = 0; j < 10; ++j) oa[j] = __shfl_xor(acc[h][j], xm, 32);
      const float nm = fmaxf(m[h], om);
      const float f1 = __expf(m[h] - nm);
      const float f2 = __expf(om - nm);
      l[h] = l[h] * f1 + ol * f2;
#pragma unroll
      for (int j = 0; j < 10; ++j) acc[h][j] = acc[h][j] * f1 + oa[j] * f2;
      m[h] = nm;
    }
  }

  // cross-wave reduction via LDS
  __shared__ float lds[8 * 4 * 4 * 12];   // [wave][part][head][m,l,acc10]
  if (kq == 0) {
#pragma unroll
    for (int h = 0; h < 4; ++h) {
      float* rec = lds + ((wave * 4 + part) * 4 + h) * 12;
      rec[0] = m[h]; rec[1] = l[h];
#pragma unroll
      for (int j = 0; j < 10; ++j) rec[2 + j] = acc[h][j];
    }
  }
  __syncthreads();

  if (tid < 16) {
    const int p2 = tid >> 2, h2 = tid & 3;
    float fm = -INFINITY, fl = 0.f, fa[10];
#pragma unroll
    for (int j = 0; j < 10; ++j) fa[j] = 0.f;
    for (int w = 0; w < 8; ++w) {
      const float* rec = lds + ((w * 4 + p2) * 4 + h2) * 12;
      const float om = rec[0], ol = rec[1];
      const float nm = fmaxf(fm, om);
      const float f1 = __expf(fm - nm);
      const float f2 = __expf(om - nm);
      fl = fl * f1 + ol * f2;
#pragma unroll
      for (int j = 0; j < 10; ++j) fa[j] = fa[j] * f1 + rec[2 + j] * f2;
      fm = nm;
    }
    const int head = kv * 4 + h2;
    float* out = partial + (size_t)((b * NHQ + head) * NCHUNK + chunk) * PREC;
    if (p2 == 0) { out[0] = fm; out[1] = fl; }
#pragma unroll
    for (int j = 0; j < 10; ++j) out[2 + p2 * 10 + j] = fa[j];
  }
}

// -------- Kernel 4: combine chunk partials + the new (Tq=1) key -----------
__global__ void finalize_kernel(const float* __restrict__ partial,
                                const float* __restrict__ qbuf,
                                const float* __restrict__ kbuf,
                                const float* __restrict__ vbuf,
                                const float* __restrict__ maskp,
                                float* __restrict__ attn_out)
{
  const int bh = blockIdx.x;       // b*NHQ + h
  const int b  = bh / NHQ;
  const int h  = bh % NHQ;
  const int kv = h >> 2;
  const int d  = threadIdx.x;
  if (d >= HD) return;
  const float scale = 0.15811388300841897f;

  float m = -INFINITY, l = 0.f, a = 0.f;
  for (int c = 0; c < NCHUNK; ++c) {
    const float* rec = partial + (size_t)(bh * NCHUNK + c) * PREC;
    const float om = rec[0], ol = rec[1], oa = rec[2 + d];
    const float nm = fmaxf(m, om);
    const float f1 = __expf(m - nm);
    const float f2 = __expf(om - nm);
    l = l * f1 + ol * f2;
    a = a * f1 + oa * f2;
    m = nm;
  }
  // new key (already RoPE'd in kbuf)
  float s = 0.f;
  const float* qv = qbuf + b * 320 + h * HD;
  const float* kn = kbuf + b * 80 + kv * HD;
#pragma unroll
  for (int j = 0; j < HD; ++j) s += qv[j] * kn[j];
  s = s * scale + maskp[b * TTOT + TPAST];
  const float nm = fmaxf(m, s);
  const float f1 = __expf(m - nm);
  const float p  = __expf(s - nm);
  l = l * f1 + p;
  a = a * f1 + p * vbuf[b * 80 + kv * HD + d];
  attn_out[b * 320 + h * HD + d] = a / l;
}

// -------- Kernel 5: output projection (WMMA) ------------------------------
__global__ void oproj_kernel(const float* __restrict__ attn_out,
                             const float* __restrict__ Wo,
                             float* __restrict__ out)
{
  const int tile = blockIdx.x;     // 40 tiles: 2 (M) x 20 (N)
  const int tm = tile & 1;
  const int tn = tile >> 1;
  float c[8];
  wmma_tile_f32(attn_out, HDIM, tm * 16, Wo, HDIM, tn * 16, HDIM, c);
  const int lane = threadIdx.x & 31;
  const int lm   = lane & 15;
  const int half = lane >> 4;
#pragma unroll
  for (int r = 0; r < 8; ++r)
    out[(tm * 16 + r + 8 * half) * HDIM + tn * 16 + lm] = c[r];
}

// -------- Host launcher ---------------------------------------------------
extern "C" void kernel_launch(void* const* d_in, const int* in_sizes, int n_in,
                              void* d_out, int out_size, void* d_ws, size_t ws_size,
                              hipStream_t stream)
{
  const float* hs    = (const float*)d_in[0];  // (32,1,320)
  const float* cosb  = (const float*)d_in[1];  // (32,1,32)
  const float* sinb  = (const float*)d_in[2];  // (32,1,32)
  const float* kc    = (const float*)d_in[3];  // (32,2,8192,40)
  const float* vc    = (const float*)d_in[4];  // (32,2,8192,40)
  const float* maskp = (const float*)d_in[5];  // (32,1,1,8193)
  const float* Wq    = (const float*)d_in[6];  // (320,320)
  const float* Wk    = (const float*)d_in[7];  // (80,320)
  const float* Wv    = (const float*)d_in[8];  // (80,320)
  const float* Wo    = (const float*)d_in[9];  // (320,320)
  float* out = (float*)d_out;                  // (32,1,320)

  float* ws       = (float*)d_ws;
  float* qbuf     = ws;                              // 32*320  = 10240
  float* kbuf     = qbuf + BB * HDIM;                // 32*80   = 2560
  float* vbuf     = kbuf + BB * NKVH * HD;           // 32*80   = 2560
  float* partial  = vbuf + BB * NKVH * HD;           // 32*8*8*42 = 86016
  float* attn_out = partial + BB * NHQ * NCHUNK * PREC; // 10240

  qkv_proj_kernel<<<60, 32, 0, stream>>>(hs, Wq, Wk, Wv, qbuf, kbuf, vbuf);
  rope_kernel<<<BB, 160, 0, stream>>>(cosb, sinb, qbuf, kbuf);
  flash_partial_kernel<<<BB * NKVH * NCHUNK, 256, 0, stream>>>(kc, vc, maskp, qbuf, partial);
  finalize_kernel<<<BB * NHQ, 64, 0, stream>>>(partial, qbuf, kbuf, vbuf, maskp, attn_out);
  oproj_kernel<<<40, 32, 0, stream>>>(attn_out, Wo, out);
}